// GCNSimple_2001454760654
// MI455X (gfx1250) — compile-verified
//
#include <hip/hip_runtime.h>

#define NNODES 100000
#define NEDGES 3200000

typedef __attribute__((ext_vector_type(2))) float v2f;
typedef __attribute__((ext_vector_type(8))) float v8f;

// ---- Pass 1: degree (self-loop seeds deg=1) --------------------------------
__global__ void gcn_init_deg(float* __restrict__ deg) {
    int v = blockIdx.x * 256 + threadIdx.x;
    if (v < NNODES) deg[v] = 1.0f;
}

__global__ void gcn_accum_deg(const int* __restrict__ col, float* __restrict__ deg) {
    int e = blockIdx.x * 256 + threadIdx.x;
    if (e < NEDGES) atomicAdd(&deg[col[e]], 1.0f);
}

__global__ void gcn_make_dinv(const float* __restrict__ deg, float* __restrict__ dinv) {
    int v = blockIdx.x * 256 + threadIdx.x;
    if (v < NNODES) dinv[v] = rsqrtf(deg[v]);   // deg >= 1 always (self-loop)
}

// ---- Pass 2: g = (x @ W) * dinv[row]; d_out seeded with self-loop term -----
// Wave-per-16x64-tile fp32 WMMA GEMM. 8 waves/block => 128 rows per block.
__global__ __launch_bounds__(256) void gcn_gemm_scale(
        const float* __restrict__ x, const float* __restrict__ W,
        const float* __restrict__ dinv, float* __restrict__ g,
        float* __restrict__ out) {
    // W repacked in LDS so each lane's two K-values are adjacent (b64 loads):
    // Wl[(k>>1)*128 + n*2 + (k&1)] = W[k*64 + n]
    __shared__ float Wl[256 * 64];
    const int tid = threadIdx.x;
    for (int i = tid; i < 256 * 64; i += 256) {
        int k = i >> 6, n = i & 63;
        Wl[(k >> 1) * 128 + (n << 1) + (k & 1)] = W[i];
    }
    __syncthreads();

    const int wave = tid >> 5;
    const int lane = tid & 31;
    const int half = lane >> 4;   // 0: lanes 0-15, 1: lanes 16-31
    const int mr   = lane & 15;
    const int m0   = blockIdx.x * 128 + wave * 16;
    if (m0 >= NNODES) return;     // wave-uniform: EXEC stays all-ones for WMMA

    const float* xrow = x + (size_t)(m0 + mr) * 256;
    const v2f*   Wp   = (const v2f*)Wl;   // pair index = (k>>1)*64 + n

    v8f acc0 = {}, acc1 = {}, acc2 = {}, acc3 = {};
    for (int k0 = 0; k0 < 256; k0 += 4) {
        const int ka = k0 + half * 2;                 // this lane's K pair
        v2f a = *(const v2f*)(xrow + ka);             // A: 16x4 fp32 frag
        const int kp = ka >> 1;
        v2f b0 = Wp[kp * 64 + mr];                    // B: 4x16 frags, 4 N-tiles
        v2f b1 = Wp[kp * 64 + 16 + mr];
        v2f b2 = Wp[kp * 64 + 32 + mr];
        v2f b3 = Wp[kp * 64 + 48 + mr];
        acc0 = __builtin_amdgcn_wmma_f32_16x16x4_f32(false, a, false, b0, (short)0, acc0, false, false);
        acc1 = __builtin_amdgcn_wmma_f32_16x16x4_f32(false, a, false, b1, (short)0, acc1, false, false);
        acc2 = __builtin_amdgcn_wmma_f32_16x16x4_f32(false, a, false, b2, (short)0, acc2, false, false);
        acc3 = __builtin_amdgcn_wmma_f32_16x16x4_f32(false, a, false, b3, (short)0, acc3, false, false);
    }

    // C/D layout: VGPR i <-> row m0 + i + 8*half, column mr (+16 per N-tile)
    #pragma unroll
    for (int i = 0; i < 8; ++i) {
        const int m = m0 + i + half * 8;
        const float s = dinv[m];
        const size_t rb = (size_t)m * 64 + mr;
        float v0 = acc0[i] * s; g[rb]      = v0; out[rb]      = v0;
        float v1 = acc1[i] * s; g[rb + 16] = v1; out[rb + 16] = v1;
        float v2 = acc2[i] * s; g[rb + 32] = v2; out[rb + 32] = v2;
        float v3 = acc3[i] * s; g[rb + 48] = v3; out[rb + 48] = v3;
    }
}

// ---- Pass 3: edge scatter (dominant, L2-resident atomics) ------------------
// 64 threads per edge -> coalesced 128B gather of g[row], contiguous atomics.
__global__ void gcn_scatter(const float* __restrict__ g,
                            const int* __restrict__ row,
                            const int* __restrict__ col,
                            float* __restrict__ out) {
    size_t t = (size_t)blockIdx.x * 256 + threadIdx.x;
    size_t e = t >> 6;
    int    n = (int)(t & 63);
    if (e < (size_t)NEDGES) {
        int r = row[e], c = col[e];
        atomicAdd(&out[(size_t)c * 64 + n], g[(size_t)r * 64 + n]);
    }
}

// ---- Pass 4: out = dinv[c]*acc + b ----------------------------------------
__global__ void gcn_finalize(const float* __restrict__ dinv,
                             const float* __restrict__ b,
                             float* __restrict__ out) {
    size_t t = (size_t)blockIdx.x * 256 + threadIdx.x;
    if (t < (size_t)NNODES * 64) {
        int v = (int)(t >> 6), n = (int)(t & 63);
        out[t] = dinv[v] * out[t] + b[n];
    }
}

extern "C" void kernel_launch(void* const* d_in, const int* in_sizes, int n_in,
                              void* d_out, int out_size, void* d_ws, size_t ws_size,
                              hipStream_t stream) {
    const float* x  = (const float*)d_in[0];
    const int*   ei = (const int*)d_in[1];     // edge_index [2, E]
    const float* W  = (const float*)d_in[2];
    const float* b  = (const float*)d_in[3];
    float* out = (float*)d_out;

    const int* row = ei;            // src
    const int* col = ei + NEDGES;   // dst

    float* deg  = (float*)d_ws;                  // N floats
    float* dinv = deg + NNODES;                  // N floats
    float* g    = dinv + NNODES;                 // N*64 floats (25.6 MB)

    gcn_init_deg  <<<(NNODES + 255) / 256, 256, 0, stream>>>(deg);
    gcn_accum_deg <<<(NEDGES + 255) / 256, 256, 0, stream>>>(col, deg);
    gcn_make_dinv <<<(NNODES + 255) / 256, 256, 0, stream>>>(deg, dinv);
    gcn_gemm_scale<<<(NNODES + 127) / 128, 256, 0, stream>>>(x, W, dinv, g, out);
    gcn_scatter   <<<(int)(((size_t)NEDGES * 64) / 256), 256, 0, stream>>>(g, row, col, out);
    gcn_finalize  <<<(NNODES * 64 + 255) / 256, 256, 0, stream>>>(dinv, b, out);
}